// Generator_66709432041924
// MI455X (gfx1250) — compile-verified
//
#include <hip/hip_runtime.h>
#include <hip/hip_bf16.h>

typedef __attribute__((ext_vector_type(2))) float v2f;
typedef __attribute__((ext_vector_type(8))) float v8f;
typedef __attribute__((ext_vector_type(4))) unsigned int u32x4;
typedef __attribute__((ext_vector_type(8))) int i32x8;
typedef __attribute__((ext_vector_type(4))) int i32x4;

// W fragment layout: pair (k, k+1) at fixed column contiguous 8B:
//   wl[(k>>1)*S2 + n*2 + (k&1)],  S2 = 290 words (bank-conflict-free halves).
#define S2 290
// A tile: row-major, TDM-padded to SA = 162 words/row (34-dword pad per 128):
//   162 mod 64 = 34 -> {34m} distinct, lane-halves (+2 words) disjoint.
#define SA 162
#define GEMM_LDS_BYTES ((64 * S2 + 128 * SA) * 4)   // 157184 B

// ---------------------------------------------------------------------------
// Out[r][n] (+)= A[r][:] @ W[:][n], 128 cols, K=128, fp32 WMMA 16x16x4.
// 256 threads = 8 waves; wave computes a 16x128 strip; block does 128 rows.
// A strip staged by the Tensor Data Mover; W staged+swizzled by ALU.
// ---------------------------------------------------------------------------
template<bool ACCUM, bool BIAS, bool RELU>
__global__ void gemm_nk128(const float* __restrict__ A,
                           const float* __restrict__ W,
                           const float* __restrict__ bias,
                           float* __restrict__ Out,
                           int nrows)
{
    extern __shared__ float lds[];
    float* wl = lds;              // 64 * S2 floats : W fragments
    float* al = lds + 64 * S2;    // 128 * SA floats : A rows (TDM, padded)

    const int tid  = threadIdx.x;
    const int wave = tid >> 5;
    const int lane = tid & 31;
    const int m    = lane & 15;   // A-row-in-frag / B,D-column-in-frag
    const int half = lane >> 4;   // k-pair selector
    const int row0 = blockIdx.x * 128;

    // ---- wave 0: kick off TDM load of the A strip (async, TENSORcnt) ----
    if (wave == 0) {
        const unsigned long long ga =
            (unsigned long long)(uintptr_t)(A + (size_t)row0 * 128);
        const unsigned lds_off = (unsigned)(uintptr_t)al;   // LDS aperture: offset in [31:0]
        const unsigned dim1 = (unsigned)(nrows - row0);     // OOB rows read as zero

        u32x4 g0;
        g0.x = 1u;                                          // count=1 (valid user D#)
        g0.y = lds_off;                                     // lds_addr (bytes)
        g0.z = (unsigned)(ga & 0xFFFFFFFFu);                // global_addr[31:0]
        g0.w = (unsigned)((ga >> 32) & 0x01FFFFFFu)         // global_addr[56:32]
             | (2u << 30);                                  // type = 2 ("image")

        i32x8 g1;
        g1[0] = (int)((2u << 16)                            // data_size = 4 bytes
              | (1u << 20)                                  // pad_enable
              | (6u << 22)                                  // pad_interval: 128 dwords
              | (33u << 25));                               // pad_amount: 34 dwords
        g1[1] = (int)((128u & 0xFFFFu) << 16);              // tensor_dim0[15:0] = 128
        g1[2] = (int)((dim1 & 0xFFFFu) << 16);              // tensor_dim0[31:16]=0 | tensor_dim1[15:0]
        g1[3] = (int)((dim1 >> 16) | (128u << 16));         // tensor_dim1[31:16] | tile_dim0 = 128
        g1[4] = (int)128u;                                  // tile_dim1 = 128 | tile_dim2 = 0
        g1[5] = (int)128u;                                  // tensor_dim0_stride[31:0] = 128
        g1[6] = 0;                                          // stride0[47:32] | stride1[15:0]
        g1[7] = 0;

        const i32x4 gz4 = {0, 0, 0, 0};                     // groups 2/3: unused (2-D)
        const i32x8 gz8 = {0, 0, 0, 0, 0, 0, 0, 0};         // trailing group (unused)
        __builtin_amdgcn_tensor_load_to_lds(g0, g1, gz4, gz4, gz8, 0);
    }

    // ---- all waves: stage W (128x128) into fragment-swizzled LDS ----
    for (int i = tid; i < 128 * 128; i += 256) {
        const int k = i >> 7, n = i & 127;
        wl[(k >> 1) * S2 + n * 2 + (k & 1)] = W[i];
    }

    if (wave == 0) __builtin_amdgcn_s_wait_tensorcnt(0);    // TDM done (wave 0)
    __syncthreads();                                        // publish LDS to all waves

    v8f acc[8];
    const v8f vzero = {0.f, 0.f, 0.f, 0.f, 0.f, 0.f, 0.f, 0.f};
#pragma unroll
    for (int t = 0; t < 8; ++t) acc[t] = vzero;

    const int arow_base = (wave * 16 + m) * SA + 2 * half;  // + ka gives frag addr
    const int m2 = m * 2;

#pragma unroll 4
    for (int kk = 0; kk < 64; kk += 2) {   // kk = k0/2 ; each step covers K += 4
        const v2f a = *(const v2f*)&al[arow_base + kk * 2];
        const float* bp = &wl[(kk + half) * S2 + m2];
#pragma unroll
        for (int t = 0; t < 8; ++t) {
            const v2f b = *(const v2f*)(bp + t * 32);       // 16 cols * 2 words
            acc[t] = __builtin_amdgcn_wmma_f32_16x16x4_f32(
                false, a, false, b, (short)0, acc[t], false, false);
        }
    }

    // Epilogue: D layout -> row = row0 + wave*16 + v + 8*half, col = t*16 + m
#pragma unroll
    for (int t = 0; t < 8; ++t) {
        const int n = t * 16 + m;
#pragma unroll
        for (int v = 0; v < 8; ++v) {
            const int r = row0 + wave * 16 + v + 8 * half;
            if (r < nrows) {
                float val = acc[t][v];
                if (ACCUM) val += Out[(size_t)r * 128 + n];
                if (BIAS)  val += bias[n];
                if (RELU)  val = fmaxf(val, 0.0f);
                Out[(size_t)r * 128 + n] = val;
            }
        }
    }
}

// ---------------------------------------------------------------------------
// Edge scatter: one wave per edge, 32 lanes x float4 = 128 floats,
// global_atomic_add_f32 into L2-resident accumulator; prefetch next gather.
// ---------------------------------------------------------------------------
__global__ void scatter_add_128(const float* __restrict__ hw,
                                const int* __restrict__ src,
                                const int* __restrict__ dst,
                                float* __restrict__ agg,
                                int nedges)
{
    const int lane   = threadIdx.x & 31;
    int e            = (int)(((size_t)blockIdx.x * blockDim.x + threadIdx.x) >> 5);
    const int stride = (gridDim.x * blockDim.x) >> 5;
    for (; e < nedges; e += stride) {
        const int s = src[e];
        const int d = dst[e];
        if (e + stride < nedges) {  // hide next gather behind this edge's atomics
            __builtin_prefetch(hw + (size_t)src[e + stride] * 128 + lane * 4, 0, 0);
        }
        const float4 v = *(const float4*)(hw + (size_t)s * 128 + lane * 4);
        float* p = agg + (size_t)d * 128 + lane * 4;
        unsafeAtomicAdd(p + 0, v.x);
        unsafeAtomicAdd(p + 1, v.y);
        unsafeAtomicAdd(p + 2, v.z);
        unsafeAtomicAdd(p + 3, v.w);
    }
}

__global__ void degree_k(const int* __restrict__ dst, float* __restrict__ deg, int nedges)
{
    const int i = blockIdx.x * blockDim.x + threadIdx.x;
    if (i < nedges) unsafeAtomicAdd(&deg[dst[i]], 1.0f);
}

__global__ void zero_k(float* __restrict__ p, size_t n)
{
    const size_t i = (size_t)blockIdx.x * blockDim.x + threadIdx.x;
    if (i < n) p[i] = 0.0f;
}

// out[i] = agg[i] * (1/max(deg,1)) + bias[col], optional relu; n = rows
template<bool RELU>
__global__ void scale_bias_k(const float* __restrict__ agg,
                             const float* __restrict__ deg,
                             const float* __restrict__ bias,
                             float* __restrict__ out, int n)
{
    const int i = blockIdx.x * blockDim.x + threadIdx.x;
    if (i < n * 128) {
        const int row = i >> 7;
        const int col = i & 127;
        const float inv = 1.0f / fmaxf(deg[row], 1.0f);
        float v = agg[i] * inv + bias[col];
        if (RELU) v = fmaxf(v, 0.0f);
        out[i] = v;
    }
}

// out[r][:] = maybe-relu(h[idx[r]][:])
template<bool RELU>
__global__ void gather_k(const float* __restrict__ h, const int* __restrict__ idx,
                         float* __restrict__ out, int mrows)
{
    const int i = blockIdx.x * blockDim.x + threadIdx.x;
    if (i < mrows * 128) {
        const int r = i >> 7;
        const int c = i & 127;
        float v = h[(size_t)idx[r] * 128 + c];
        if (RELU) v = fmaxf(v, 0.0f);
        out[i] = v;
    }
}

static inline int cdiv(int a, int b) { return (a + b - 1) / b; }

extern "C" void kernel_launch(void* const* d_in, const int* in_sizes, int n_in,
                              void* d_out, int out_size, void* d_ws, size_t ws_size,
                              hipStream_t stream)
{
    const float* feat   = (const float*)d_in[0];
    const int*   src    = (const int*)d_in[1];
    const int*   dst    = (const int*)d_in[2];
    const int*   masked = (const int*)d_in[3];
    const float* W1     = (const float*)d_in[4];
    const float* b1     = (const float*)d_in[5];
    const float* W2     = (const float*)d_in[6];
    const float* b2     = (const float*)d_in[7];
    const float* Wm1    = (const float*)d_in[8];
    const float* bm1    = (const float*)d_in[9];
    const float* Wm2    = (const float*)d_in[10];
    const float* bm2    = (const float*)d_in[11];
    const float* Wm3    = (const float*)d_in[12];

    const int N = in_sizes[0] / 128;
    const int E = in_sizes[1];
    const int M = in_sizes[3];

    float* ws   = (float*)d_ws;
    float* deg  = ws;  ws += N;
    float* buf0 = ws;  ws += (size_t)N * 128;   // GEMM output (hw)
    float* buf1 = ws;  ws += (size_t)N * 128;   // scatter accumulator
    float* buf2 = ws;  ws += (size_t)N * 128;   // activations h
    float* hm   = ws;  ws += (size_t)M * 128;
    float* fm   = ws;  ws += (size_t)M * 128;
    float* t1   = ws;  ws += (size_t)M * 128;
    float* t2   = ws;  ws += (size_t)M * 128;
    float* outp = (float*)d_out;

    const size_t n128 = (size_t)N * 128;
    const int    zb   = (int)((n128 + 255) / 256);

    // degrees
    zero_k<<<cdiv(N, 256), 256, 0, stream>>>(deg, (size_t)N);
    degree_k<<<cdiv(E, 256), 256, 0, stream>>>(dst, deg, E);

    // ---- GCN layer 1: h1 = relu(scatter(feat@W1) * inv_deg + b1)
    gemm_nk128<false, false, false><<<cdiv(N, 128), 256, GEMM_LDS_BYTES, stream>>>(feat, W1, nullptr, buf0, N);
    zero_k<<<zb, 256, 0, stream>>>(buf1, n128);
    scatter_add_128<<<2048, 256, 0, stream>>>(buf0, src, dst, buf1, E);
    scale_bias_k<true><<<cdiv(N * 128, 256), 256, 0, stream>>>(buf1, deg, b1, buf2, N);

    // ---- GCN layer 2: h2 = scatter(h1@W2) * inv_deg + b2
    gemm_nk128<false, false, false><<<cdiv(N, 128), 256, GEMM_LDS_BYTES, stream>>>(buf2, W2, nullptr, buf0, N);
    zero_k<<<zb, 256, 0, stream>>>(buf1, n128);
    scatter_add_128<<<2048, 256, 0, stream>>>(buf0, src, dst, buf1, E);
    scale_bias_k<false><<<cdiv(N * 128, 256), 256, 0, stream>>>(buf1, deg, b2, buf2, N);

    // ---- gather masked rows
    gather_k<true><<<cdiv(M * 128, 256), 256, 0, stream>>>(buf2, masked, hm, M);
    gather_k<false><<<cdiv(M * 128, 256), 256, 0, stream>>>(feat, masked, fm, M);

    // ---- MLP: [hm|fm]@Wm1 split into two K=128 GEMMs (accumulate), bias+relu on 2nd
    gemm_nk128<false, false, false><<<cdiv(M, 128), 256, GEMM_LDS_BYTES, stream>>>(hm, Wm1, nullptr, t1, M);
    gemm_nk128<true, true, true><<<cdiv(M, 128), 256, GEMM_LDS_BYTES, stream>>>(fm, Wm1 + 128 * 128, bm1, t1, M);
    gemm_nk128<false, true, true><<<cdiv(M, 128), 256, GEMM_LDS_BYTES, stream>>>(t1, Wm2, bm2, t2, M);
    gemm_nk128<false, false, false><<<cdiv(M, 128), 256, GEMM_LDS_BYTES, stream>>>(t2, Wm3, nullptr, outp, M);
}